// SoftDTW_69458211110933
// MI455X (gfx1250) — compile-verified
//
#include <hip/hip_runtime.h>
#include <hip/hip_bf16.h>

#define BIG 1.0e10f

typedef float v2f __attribute__((ext_vector_type(2)));
typedef float v8f __attribute__((ext_vector_type(8)));
typedef int i32x4 __attribute__((vector_size(16)));
typedef __attribute__((address_space(1))) i32x4 gi32x4;   // global int32x4
typedef __attribute__((address_space(3))) i32x4 li32x4;   // LDS int32x4

// ---- CDNA5 async global->LDS copy (ASYNCcnt path), with safe fallback -----
#if defined(__has_builtin)
#if __has_builtin(__builtin_amdgcn_global_load_async_to_lds_b128)
#define HAVE_ASYNC_LDS 1
#endif
#endif

#ifdef HAVE_ASYNC_LDS
#define ASYNC_B128(gsrc, ldst) \
  __builtin_amdgcn_global_load_async_to_lds_b128((gi32x4*)(gsrc), (li32x4*)(ldst), 0, 0)
#else
#define ASYNC_B128(gsrc, ldst) \
  do { *(float4*)(ldst) = *(const float4*)(gsrc); } while (0)
#endif

#if defined(__has_builtin) && __has_builtin(__builtin_amdgcn_s_wait_asynccnt)
#define WAIT_ASYNC() __builtin_amdgcn_s_wait_asynccnt(0)
#else
#define WAIT_ASYNC() asm volatile("s_wait_asynccnt 0x0" ::: "memory")
#endif

// ---------------------------------------------------------------------------
// Kernel 0: row norms  ||x_i||^2, ||y_j||^2   (32*512 each)
// ---------------------------------------------------------------------------
__global__ __launch_bounds__(256) void softdtw_norms(
    const float* __restrict__ x, const float* __restrict__ y,
    float* __restrict__ x2, float* __restrict__ y2) {
  int t = blockIdx.x * 256 + threadIdx.x;          // 0 .. 32767
  const float* src;
  float* dst;
  int idx;
  if (t < 16384) { src = x; dst = x2; idx = t; }
  else           { src = y; dst = y2; idx = t - 16384; }
  const float4* p = (const float4*)(src + (size_t)idx * 64);
  float s = 0.0f;
#pragma unroll
  for (int k = 0; k < 16; ++k) {
    float4 v = p[k];
    s += v.x * v.x + v.y * v.y + v.z * v.z + v.w * v.w;
  }
  dst[idx] = s;
}

// ---------------------------------------------------------------------------
// Kernel 1: batched distance matrix. One block = one 64x64 tile of D[b].
// Stage 64x64 x-slab and 64x64 y-slab (each a contiguous 16KB region of
// global) into LDS with GLOBAL_LOAD_ASYNC_TO_LDS_B128, fence with
// s_wait_asynccnt + barrier, then 8 waves x 2 sub-tiles of
// V_WMMA_F32_16X16X4_F32 fed by conflict-free ds_load_b64 (row stride 68).
// ---------------------------------------------------------------------------
#define LSTR 68   // LDS row stride in floats: banks = (4*row + col) % 64

__global__ __launch_bounds__(256) void softdtw_gemm(
    const float* __restrict__ x, const float* __restrict__ y,
    const float* __restrict__ x2, const float* __restrict__ y2,
    float* __restrict__ Dmat) {
  __shared__ float xs[64 * LSTR];
  __shared__ float ys[64 * LSTR];

  const int b  = blockIdx.x >> 6;          // 64 tiles per batch (8x8)
  const int tt = blockIdx.x & 63;
  const int i0 = (tt >> 3) << 6;           // 64-row slab origin
  const int j0 = (tt & 7) << 6;            // 64-col slab origin

  const float* gx = x + ((size_t)b << 15) + (size_t)i0 * 64;  // 16KB contiguous
  const float* gy = y + ((size_t)b << 15) + (size_t)j0 * 64;

  const int t = threadIdx.x;               // 0..255
#pragma unroll
  for (int c = t; c < 1024; c += 256) {    // 1024 x 16B chunks per slab
    const int row = c >> 4;
    const int off = (c & 15) << 2;         // float index within row
    ASYNC_B128(gx + (c << 2), &xs[row * LSTR + off]);
    ASYNC_B128(gy + (c << 2), &ys[row * LSTR + off]);
  }
  WAIT_ASYNC();
  __syncthreads();

  const int w    = t >> 5;                 // wave 0..7
  const int lane = t & 31;
  const int half = lane >> 4;              // K pair select
  const int mn   = lane & 15;              // A-row / B-col / C-col
  const int ti   = w >> 1;                 // sub-tile row 0..3
  const int tj0  = (w & 1) << 1;           // sub-tile cols {tj0, tj0+1}

  const float* ax  = &xs[(ti * 16 + mn) * LSTR + (half << 1)];
  const float* by0 = &ys[(tj0 * 16 + mn) * LSTR + (half << 1)];
  const float* by1 = by0 + 16 * LSTR;

  v8f acc0 = {}, acc1 = {};
#pragma unroll
  for (int k = 0; k < 64; k += 4) {
    v2f a  = *(const v2f*)(ax + k);        // ds_load_b64, shared by both tiles
    v2f b0 = *(const v2f*)(by0 + k);
    v2f b1 = *(const v2f*)(by1 + k);
    acc0 = __builtin_amdgcn_wmma_f32_16x16x4_f32(
        false, a, false, b0, (short)0, acc0, false, false);
    acc1 = __builtin_amdgcn_wmma_f32_16x16x4_f32(
        false, a, false, b1, (short)0, acc1, false, false);
  }

  float* Db = Dmat + ((size_t)b << 18);
  const float* xn = x2 + (b << 9) + i0 + ti * 16 + (half << 3);
#pragma unroll
  for (int r = 0; r < 2; ++r) {
    const v8f acc = r ? acc1 : acc0;
    const int jg  = j0 + (tj0 + r) * 16 + mn;
    const float yn = y2[(b << 9) + jg];
#pragma unroll
    for (int v = 0; v < 8; ++v) {
      const int ig = i0 + ti * 16 + (half << 3) + v;
      Db[(size_t)ig * 512 + jg] = xn[v] + yn - 2.0f * acc[v];
    }
  }
}

// ---------------------------------------------------------------------------
// Kernel 2: anti-diagonal Soft-DTW DP. One block (512 threads, 16 waves)
// per batch; rotating 3 diagonals in LDS; thread t streams D row t (L2-hot).
// ---------------------------------------------------------------------------
__global__ __launch_bounds__(512) void softdtw_dp(
    const float* __restrict__ Dmat, float* __restrict__ out) {
  __shared__ float diag[3][516];
  const int t = threadIdx.x;              // 0..511
  const int b = blockIdx.x;               // batch
  const int i = t + 1;                    // table row 1..512
  const float* drow = Dmat + ((size_t)b << 18) + (size_t)(i - 1) * 512;

  if (t == 0) { diag[0][0] = 0.0f; diag[1][0] = BIG; }
  diag[0][i] = BIG;
  diag[1][i] = BIG;
  __syncthreads();

  int p0 = 0, p1 = 1, p2 = 2;             // k-2, k-1, k buffer indices
  for (int k = 2; k <= 1024; ++k) {
    const int j = k - i;
    const float r0 = diag[p0][i - 1];     // R[i-1, j-1]
    const float r1 = diag[p1][i - 1];     // R[i-1, j]
    const float r2 = diag[p1][i];         // R[i,   j-1]
    float val = BIG;
    if (j >= 1 && j <= 512) {
      const float m = fminf(r0, fminf(r1, r2));
      const float s = __expf(m - r0) + __expf(m - r1) + __expf(m - r2);
      val = drow[j - 1] + (m - __logf(s));
    }
    if (j >= 1 && j <= 496)
      __builtin_prefetch(drow + j - 1 + 16, 0, 1);   // global_prefetch_b8
    __syncthreads();
    diag[p2][i] = val;
    if (t == 0) diag[p2][0] = BIG;
    __syncthreads();
    const int tmp = p0; p0 = p1; p1 = p2; p2 = tmp;
  }
  if (t == 511) out[b] = diag[p1][512];   // R[N, M]
}

// ---------------------------------------------------------------------------
extern "C" void kernel_launch(void* const* d_in, const int* in_sizes, int n_in,
                              void* d_out, int out_size, void* d_ws, size_t ws_size,
                              hipStream_t stream) {
  const float* x = (const float*)d_in[0];   // (32, 512, 64) f32
  const float* y = (const float*)d_in[1];   // (32, 512, 64) f32
  float* outp = (float*)d_out;              // (32,) f32

  float* Dmat = (float*)d_ws;               // 32*512*512 f32 = 33.5 MB
  float* x2   = Dmat + ((size_t)32 << 18);  // 32*512 f32
  float* y2   = x2 + (32 << 9);             // 32*512 f32

  softdtw_norms<<<128, 256, 0, stream>>>(x, y, x2, y2);
  softdtw_gemm<<<2048, 256, 0, stream>>>(x, y, x2, y2, Dmat);  // 32 * 8*8 tiles
  softdtw_dp<<<32, 512, 0, stream>>>(Dmat, outp);
}